// YoloXTrainingModel_34127810134554
// MI455X (gfx1250) — compile-verified
//
#include <hip/hip_runtime.h>
#include <math.h>

typedef float v2f __attribute__((ext_vector_type(2)));
typedef float v8f __attribute__((ext_vector_type(8)));

#define NB 32
#define NL 50
#define NC 80
#define NA 3
#define TOTAL 14175           // 60*60*3 + 30*30*3 + 15*15*3
#define BLK 128
#define IMGF 480.0f
#define OBJ_SCALE 5.0f
#define NOOBJ_SCALE 1.0f
#define IOU_IGNORE 0.6f

#if defined(__AMDGCN__) && \
    __has_builtin(__builtin_amdgcn_global_load_async_to_lds_b32) && \
    __has_builtin(__builtin_amdgcn_global_store_async_from_lds_b32) && \
    __has_builtin(__builtin_amdgcn_s_wait_asynccnt)
#define USE_ASYNC_LDS 1
#else
#define USE_ASYNC_LDS 0
#endif

#define AS_GLOBAL __attribute__((address_space(1)))
#define AS_LOCAL  __attribute__((address_space(3)))

// Per-GT record in LDS (16 floats):
// [0..3] tmin/tmax (normalized)  [4] tarea  [5] cell_i  [6] cell_j
// [7] best anchor k  [8] valid  [9..12] adj targets  [13] class id
__global__ __launch_bounds__(BLK) void yolo_level_kernel(
    const float* __restrict__ preds, const float* __restrict__ gt,
    const float* __restrict__ anchors, int H, int lvl_off,
    float* __restrict__ loss, float* __restrict__ boxes,
    float* __restrict__ scores)
{
  const int W = H;
  const float Wf = (float)W;
  const int b = blockIdx.y;
  const int cellsA = H * W * NA;
  const int blk0 = blockIdx.x * BLK;
  int remain = cellsA - blk0; if (remain > BLK) remain = BLK;

  __shared__ float spred[BLK * 85];   // staged preds, later reused for outputs
  __shared__ float sg[NL][16];
  __shared__ float sAcc;
  if (threadIdx.x == 0) sAcc = 0.0f;

  // ---- stage GT metadata (tiny; plain loads) ----
  if (threadIdx.x < NL) {
    const int l = threadIdx.x;
    const float* g = gt + (b * NL + l) * 5;
    const float gx = g[0], gy = g[1], gw = g[2], gh = g[3], gcls = g[4];
    const float bx = gx * Wf, by = gy * Wf, bw = gw * Wf, bh = gh * Wf;
    const float valid = (bw > 0.0f) ? 1.0f : 0.0f;
    const float fj = fminf(fmaxf(floorf(bx), 0.0f), Wf - 1.0f);
    const float fi = fminf(fmaxf(floorf(by), 0.0f), Wf - 1.0f);
    float bestiou = -1.0f; int k = 0;            // argmax, first max wins
    for (int a = 0; a < NA; ++a) {
      const float aw = anchors[2 * a] * Wf, ah = anchors[2 * a + 1] * Wf;
      const float inter = fminf(bw, aw) * fminf(bh, ah);
      const float iou = inter / (bw * bh + aw * ah - inter + 1e-9f);
      if (iou > bestiou) { bestiou = iou; k = a; }
    }
    const float awk = anchors[2 * k] * Wf, ahk = anchors[2 * k + 1] * Wf;
    const float sw = (valid > 0.0f) ? bw : 1.0f;
    const float sh = (valid > 0.0f) ? bh : 1.0f;
    sg[l][0]  = gx - 0.5f * gw;  sg[l][1]  = gy - 0.5f * gh;
    sg[l][2]  = gx + 0.5f * gw;  sg[l][3]  = gy + 0.5f * gh;
    sg[l][4]  = gw * gh;
    sg[l][5]  = fi;  sg[l][6] = fj;  sg[l][7] = (float)k;  sg[l][8] = valid;
    sg[l][9]  = bx - fj;         sg[l][10] = by - fi;
    sg[l][11] = __logf(sw / awk); sg[l][12] = __logf(sh / ahk);
    sg[l][13] = gcls;
  }

  // ---- stage this block's contiguous prediction span into LDS ----
  const float* gbase = preds + (size_t)b * cellsA * 85 + (size_t)blk0 * 85;
  const int spanDw = remain * 85;
#if USE_ASYNC_LDS
  for (int i = threadIdx.x; i < spanDw; i += BLK)
    __builtin_amdgcn_global_load_async_to_lds_b32(
        (AS_GLOBAL int*)(gbase + i), (AS_LOCAL int*)&spred[i], 0, 0);
  __builtin_amdgcn_s_wait_asynccnt(0);
#else
  for (int i = threadIdx.x; i < spanDw; i += BLK) spred[i] = gbase[i];
#endif
  __syncthreads();

  // ---- per cell-anchor compute (from LDS) ----
  const int idx = blk0 + threadIdx.x;
  const bool inb = idx < cellsA;
  const int ci = inb ? idx : 0;
  const int a    = ci % NA;
  const int cell = ci / NA;
  const int x    = cell % W;
  const int y    = cell / W;

  float* sp = &spred[threadIdx.x * 85];     // own slice (garbage if !inb: ok)
  const float f0 = sp[0], f1 = sp[1], f2 = sp[2], f3 = sp[3], f4 = sp[4];
  const float sigx = 1.0f / (1.0f + __expf(-f0));
  const float sigy = 1.0f / (1.0f + __expf(-f1));
  const float conf = 1.0f / (1.0f + __expf(-f4));
  const float aw = anchors[2 * a], ah = anchors[2 * a + 1];  // normalized
  const float pw = __expf(f2) * aw, ph = __expf(f3) * ah;    // anc/conv == anchors
  const float px = (sigx + (float)x) / Wf;
  const float py = (sigy + (float)y) / Wf;
  const float pminx = px - 0.5f * pw, pmaxx = px + 0.5f * pw;
  const float pminy = py - 0.5f * ph, pmaxy = py + 0.5f * ph;
  const float parea = pw * ph;

  // best IoU over GTs + inline detector-mask reconstruction
  float best = 0.0f; int match = -1;
  const float fy = (float)y, fx = (float)x, fa = (float)a;
  #pragma unroll 5
  for (int l = 0; l < NL; ++l) {
    float wi = fminf(pmaxx, sg[l][2]) - fmaxf(pminx, sg[l][0]);
    float hi = fminf(pmaxy, sg[l][3]) - fmaxf(pminy, sg[l][1]);
    wi = fmaxf(wi, 0.0f); hi = fmaxf(hi, 0.0f);
    const float inter = wi * hi;
    const float iou = inter / (parea + sg[l][4] - inter + 1e-9f);
    best = fmaxf(best, iou);
    if (sg[l][8] > 0.0f && sg[l][5] == fy && sg[l][6] == fx && sg[l][7] == fa)
      match = l;  // last writer wins (scatter-set semantics)
  }
  const float mask = (match >= 0) ? 1.0f : 0.0f;
  const float objdet = (best > IOU_IGNORE) ? 1.0f : 0.0f;

  float lpart = NOOBJ_SCALE * (1.0f - objdet) * (1.0f - mask) * conf * conf
              + OBJ_SCALE * mask * (1.0f - conf) * (1.0f - conf);
  if (match >= 0) {
    const float dx = sg[match][9]  - sigx;
    const float dy = sg[match][10] - sigy;
    const float dw = sg[match][11] - f2;
    const float dh = sg[match][12] - f3;
    lpart += dx * dx + dy * dy + dw * dw + dh * dh;
  }

  // softmax over 80 classes from LDS; pass 3 overwrites logits with scores
  float mx = -1e30f;
  for (int c = 0; c < NC; ++c) mx = fmaxf(mx, sp[5 + c]);
  float s = 0.0f;
  for (int c = 0; c < NC; ++c) s += __expf(sp[5 + c] - mx);
  const float inv = 1.0f / s;
  const int tcls = (match >= 0) ? (int)sg[match][13] : -1;
  float clsl = 0.0f;
  for (int c = 0; c < NC; ++c) {
    const float lg = sp[5 + c];
    const float p = __expf(lg - mx) * inv;
    sp[5 + c] = conf * p;                       // score -> LDS (logit now dead)
    const float d = ((c == tcls) ? 1.0f : 0.0f) - p;
    clsl += d * d;
  }
  if (match >= 0) lpart += clsl;
  if (!inb) lpart = 0.0f;

  // decoded box -> LDS slots 0..3 (f0..f3 now dead)
  sp[0] = pminx * IMGF; sp[1] = pminy * IMGF;
  sp[2] = pmaxx * IMGF; sp[3] = pmaxy * IMGF;

  // ---- wave loss reduction on the matrix unit ----
  // D = A(16x4 partials) x ones(4x16): row m = partial[m] + partial[m+16];
  // lanes <16 hold rows 0..7, lanes >=16 rows 8..15 -> lanes 0 & 16 own halves.
  // EXEC all-ones here (no early exits anywhere in the kernel).
  v2f av; av.x = lpart; av.y = 0.0f;
  v2f bv; bv.x = 1.0f;  bv.y = 1.0f;
  v8f cv = {};
  v8f dv = __builtin_amdgcn_wmma_f32_16x16x4_f32(
      false, av, false, bv, (short)0, cv, false, false);
  const float wsum = dv[0] + dv[1] + dv[2] + dv[3] + dv[4] + dv[5] + dv[6] + dv[7];
  const int lane = threadIdx.x & 31;
  if (lane == 0 || lane == 16) atomicAdd(&sAcc, wsum);
  __syncthreads();   // sAcc done + all output slices in LDS
  if (threadIdx.x == 0) atomicAdd(loss, 0.5f * sAcc);

  // ---- coalesced LDS -> global output streaming ----
  float* gsc = scores + ((size_t)b * TOTAL + lvl_off + blk0) * NC;
  float* gbx = boxes  + ((size_t)b * TOTAL + lvl_off + blk0) * 4;
  const int nsc = remain * NC;
  const int nbx = remain * 4;
#if USE_ASYNC_LDS
  for (int g = threadIdx.x; g < nsc; g += BLK) {
    const int t = g / NC, c = g - t * NC;
    __builtin_amdgcn_global_store_async_from_lds_b32(
        (AS_GLOBAL int*)(gsc + g), (AS_LOCAL int*)&spred[t * 85 + 5 + c], 0, 0);
  }
  for (int g = threadIdx.x; g < nbx; g += BLK) {
    const int t = g >> 2, c = g & 3;
    __builtin_amdgcn_global_store_async_from_lds_b32(
        (AS_GLOBAL int*)(gbx + g), (AS_LOCAL int*)&spred[t * 85 + c], 0, 0);
  }
  __builtin_amdgcn_s_wait_asynccnt(0);
#else
  for (int g = threadIdx.x; g < nsc; g += BLK) {
    const int t = g / NC, c = g - t * NC;
    gsc[g] = spred[t * 85 + 5 + c];
  }
  for (int g = threadIdx.x; g < nbx; g += BLK) {
    const int t = g >> 2, c = g & 3;
    gbx[g] = spred[t * 85 + c];
  }
#endif
}

extern "C" void kernel_launch(void* const* d_in, const int* in_sizes, int n_in,
                              void* d_out, int out_size, void* d_ws, size_t ws_size,
                              hipStream_t stream) {
  (void)in_sizes; (void)n_in; (void)out_size; (void)d_ws; (void)ws_size;
  const float* p0      = (const float*)d_in[0]; // (32,60,60,255)
  const float* p1      = (const float*)d_in[1]; // (32,30,30,255)
  const float* p2      = (const float*)d_in[2]; // (32,15,15,255)
  const float* gt      = (const float*)d_in[3]; // (32,50,5)
  const float* anchors = (const float*)d_in[4]; // (3,2)

  float* out    = (float*)d_out;
  float* loss   = out;                                   // [1]
  float* boxes  = out + 1;                               // (32,14175,4)
  float* scores = out + 1 + (size_t)NB * TOTAL * 4;      // (32,14175,80)

  (void)hipMemsetAsync(loss, 0, sizeof(float), stream);

  const dim3 blk(BLK, 1, 1);
  struct Lvl { const float* p; int H; int off; };
  const Lvl lvls[3] = { {p0, 60, 0}, {p1, 30, 10800}, {p2, 15, 13500} };
  for (int i = 0; i < 3; ++i) {
    const int cellsA = lvls[i].H * lvls[i].H * NA;
    const dim3 grd((cellsA + BLK - 1) / BLK, NB, 1);
    hipLaunchKernelGGL(yolo_level_kernel, grd, blk, 0, stream,
                       lvls[i].p, gt, anchors, lvls[i].H, lvls[i].off,
                       loss, boxes, scores);
  }
}